// SeesawLossWithLogits_16621523436027
// MI455X (gfx1250) — compile-verified
//
#include <hip/hip_runtime.h>
#include <hip/hip_bf16.h>
#include <math.h>

#define NUM_CLASSES 8192
#define BATCH 256
#define SEESAW_EPS 1e-6f

typedef __attribute__((ext_vector_type(2))) float v2f;
typedef __attribute__((ext_vector_type(8))) float v8f;

__global__ void seesaw_init_out(float* out) {
    if (threadIdx.x == 0) out[0] = 0.0f;
}

// One workgroup per batch row; 256 threads = 8 waves (wave32).
__global__ __launch_bounds__(256) void seesaw_loss_kernel(
    const float* __restrict__ logits,   // [B, C]
    const float* __restrict__ targets,  // [B, C] one-hot
    const float* __restrict__ s,        // [C, C] row-major, s[i,i] == 1
    float* __restrict__ out)            // [1] mean loss
{
    const int b   = blockIdx.x;
    const int tid = threadIdx.x;
    const float* lrow = logits  + (size_t)b * NUM_CLASSES;
    const float* trow = targets + (size_t)b * NUM_CLASSES;

    __shared__ float red[256];
    __shared__ int   lb_sh;
    __shared__ float wave_dot[8];

    if (tid == 0) lb_sh = 0;
    __syncthreads();

    // ---- Pass 1: row max + label index, float4 (16B) coalesced scan.
    float m = -3.402823466e38f;
#pragma unroll
    for (int k = 0; k < 8; ++k) {
        const int j = k * 1024 + tid * 4;
        const float4 lv = *reinterpret_cast<const float4*>(lrow + j);
        const float4 tv = *reinterpret_cast<const float4*>(trow + j);
        m = fmaxf(m, fmaxf(fmaxf(lv.x, lv.y), fmaxf(lv.z, lv.w)));
        if (tv.x > 0.5f) lb_sh = j;
        if (tv.y > 0.5f) lb_sh = j + 1;
        if (tv.z > 0.5f) lb_sh = j + 2;
        if (tv.w > 0.5f) lb_sh = j + 3;
    }
    red[tid] = m;
    __syncthreads();
    for (int off = 128; off > 0; off >>= 1) {
        if (tid < off) red[tid] = fmaxf(red[tid], red[tid + off]);
        __syncthreads();
    }
    const float mx = red[0];
    const int   lb = lb_sh;
    __syncthreads();

    // ---- Pass 2: UNMASKED dot_full = sum_j exp(l[j]-mx) * s[lb,j]
    // via V_WMMA_F32_16X16X4_F32 diagonals; the j==lb term is subtracted
    // afterwards (dot = dot_full - expl_lb * s[lb,lb]), so the inner loop
    // has NO compares/branches: EXEC stays all-1s as WMMA requires.
    //
    // A[m,k] = w[base + 4m + k], B[k,n] = r[base + 4n + k]
    //   -> C[m,m] = 4-elem partial dot; 64 elems consumed per WMMA.
    // ISA layout (16x4 f32): lane = M; VGPR0 holds K={0|2} by lane half,
    // VGPR1 holds K={1|3}; symmetric for B -> per-lane A and B use the SAME
    // consecutive index pair -> one global_load_b64 each.
    const int lane = tid & 31;
    const int wave = tid >> 5;
    const int mrow = lane & 15;
    const int hi   = lane >> 4;

    const float* __restrict__ srow = s + (size_t)lb * NUM_CLASSES;

    v8f acc0 = {};
    v8f acc1 = {};
    const int base0 = wave * 1024 + 4 * mrow + 2 * hi;  // even -> 8B aligned
#pragma unroll 2
    for (int it = 0; it < 8; ++it) {
        const int idxA = base0 + it * 128;
        const int idxB = idxA + 64;

        // prefetch ahead in the gathered s row (global_prefetch_b8)
        __builtin_prefetch(srow + idxA + 128, 0, 3);

        const float2 lvA = *reinterpret_cast<const float2*>(lrow + idxA);
        const float2 svA = *reinterpret_cast<const float2*>(srow + idxA);
        const float2 lvB = *reinterpret_cast<const float2*>(lrow + idxB);
        const float2 svB = *reinterpret_cast<const float2*>(srow + idxB);

        v2f a0;  a0[0] = __expf(lvA.x - mx);  a0[1] = __expf(lvA.y - mx);
        v2f b0;  b0[0] = svA.x;               b0[1] = svA.y;
        v2f a1;  a1[0] = __expf(lvB.x - mx);  a1[1] = __expf(lvB.y - mx);
        v2f b1;  b1[0] = svB.x;               b1[1] = svB.y;

        // 8 args: (neg_a, A, neg_b, B, c_mod, C, reuse_a, reuse_b)
        acc0 = __builtin_amdgcn_wmma_f32_16x16x4_f32(
            false, a0, false, b0, (short)0, acc0, false, false);
        acc1 = __builtin_amdgcn_wmma_f32_16x16x4_f32(
            false, a1, false, b1, (short)0, acc1, false, false);
    }
    const v8f acc = acc0 + acc1;

    // Diagonal extraction: C[m,m] lives at (VGPR=m, lane=m) for m<8 and
    // (VGPR=m-8, lane=m+16) for m>=8  -> lanes 0-7 use acc[lane],
    // lanes 24-31 use acc[lane-24], others contribute 0.
    const int sel = (lane < 8) ? lane : ((lane >= 24) ? (lane - 24) : -1);
    float diag = 0.0f;
#pragma unroll
    for (int i = 0; i < 8; ++i) diag = (sel == i) ? acc[i] : diag;

    // wave32 shuffle reduction
#pragma unroll
    for (int o = 16; o > 0; o >>= 1) diag += __shfl_xor(diag, o, 32);
    if (lane == 0) wave_dot[wave] = diag;
    __syncthreads();

    if (tid == 0) {
        float dot_full = 0.0f;
#pragma unroll
        for (int i = 0; i < 8; ++i) dot_full += wave_dot[i];
        const float expl_lb = __expf(lrow[lb] - mx);
        const float s_ll    = srow[lb];                       // == 1.0f by construction
        const float dot     = dot_full - expl_lb * s_ll;      // remove j==lb term
        const float denom   = dot + expl_lb;
        const float sigma   = expl_lb / (denom + SEESAW_EPS);
        const float loss    = -__logf(sigma + SEESAW_EPS);
        atomicAdd(out, loss * (1.0f / (float)BATCH));
    }
}

extern "C" void kernel_launch(void* const* d_in, const int* in_sizes, int n_in,
                              void* d_out, int out_size, void* d_ws, size_t ws_size,
                              hipStream_t stream) {
    (void)in_sizes; (void)n_in; (void)out_size; (void)d_ws; (void)ws_size;
    const float* logits  = (const float*)d_in[0];
    const float* targets = (const float*)d_in[1];
    const float* s       = (const float*)d_in[2];
    float* out = (float*)d_out;

    seesaw_init_out<<<1, 32, 0, stream>>>(out);
    seesaw_loss_kernel<<<BATCH, 256, 0, stream>>>(logits, targets, s, out);
}